// MultilabelLoss_3882650436522
// MI455X (gfx1250) — compile-verified
//
#include <hip/hip_runtime.h>
#include <hip/hip_bf16.h>

// MultilabelLoss for MI455X (gfx1250, wave32).
// Single streaming pass over the 128MB activation matrix (HBM-bound, ~5.5us
// floor at 23.3 TB/s). Segment sums computed as one-hot x X matmuls with
// V_WMMA_F32_16X16X4_F32; std via sumsq - n*mean^2 identity (one pass, not two).

typedef __attribute__((ext_vector_type(2))) float v2f;
typedef __attribute__((ext_vector_type(8))) float v8f;

#define VIEWS 6
#define ROW_CHUNKS 16

// ---------------- workspace zeroing ----------------
__global__ void ml_zero_kernel(float* __restrict__ p, int n) {
    int i = blockIdx.x * blockDim.x + threadIdx.x;
    if (i < n) p[i] = 0.0f;
}

// ---------------- main streaming pass ----------------
// One wave per (16-column tile, row chunk). Accumulates per-view sum and
// sum-of-squares for its 16 columns via WMMA with a one-hot A matrix,
// then atomically folds the 6 valid accumulator rows into global partials.
__global__ void ml_seg_wmma_kernel(const float* __restrict__ x,
                                   const int* __restrict__ views,
                                   float* __restrict__ sums,   // (VIEWS, C)
                                   float* __restrict__ sumsq,  // (VIEWS, C)
                                   int N, int C) {
    const int gtid = blockIdx.x * blockDim.x + threadIdx.x;
    const int wave = gtid >> 5;
    const int lane = gtid & 31;

    const int col_tiles = C >> 4;           // 16 columns per tile
    const int col_tile  = wave % col_tiles;
    const int chunk     = wave / col_tiles; // grid sized exactly: chunk < ROW_CHUNKS

    int rows_per_chunk = (N + ROW_CHUNKS - 1) / ROW_CHUNKS;
    rows_per_chunk = (rows_per_chunk + 15) & ~15;   // multiple of 16 (iter granule)
    const int row_begin = chunk * rows_per_chunk;
    int row_end = row_begin + rows_per_chunk;
    if (row_end > N) row_end = N;

    const int m   = lane & 15;              // A-matrix M index (view id, 6..15 are zero pad)
    const int klo = (lane < 16) ? 0 : 2;    // K pair held by this half-wave
    const int col = (col_tile << 4) + m;    // B/C/D N index = lane % 16

    v8f accS = {};   // per-view sum        (rows M=0..5 of D are the live data)
    v8f accQ = {};   // per-view sum of x^2

    for (int r0 = row_begin; r0 < row_end; r0 += 16) {
#pragma unroll
        for (int s = 0; s < 4; ++s) {       // 4 k-steps of K=4 -> 16 rows per iter
            const int ra = r0 + 4 * s + klo;
            const int rb = ra + 1;
            const int sa = (ra < N) ? ra : (N - 1);   // clamp keeps loads in-bounds
            const int sb = (rb < N) ? rb : (N - 1);

            const int va = views[sa];
            const int vb = views[sb];
            v2f a;
            a.x = ((ra < N) && (va == m)) ? 1.0f : 0.0f;  // one-hot A (zero on pad/OOB)
            a.y = ((rb < N) && (vb == m)) ? 1.0f : 0.0f;

            v2f b;
            b.x = x[(size_t)sa * (size_t)C + (size_t)col];
            b.y = x[(size_t)sb * (size_t)C + (size_t)col];
            v2f bq;
            bq.x = b.x * b.x;
            bq.y = b.y * b.y;

            // D = A(one-hot 16x4) * B(4x16) + C  -> v_wmma_f32_16x16x4_f32
            accS = __builtin_amdgcn_wmma_f32_16x16x4_f32(
                false, a, false, b,  (short)0, accS, false, false);
            accQ = __builtin_amdgcn_wmma_f32_16x16x4_f32(
                false, a, false, bq, (short)0, accQ, false, false);
        }
    }

    // C/D layout: VGPR j, lanes 0-15 hold row M=j, N=lane. Views 0..5 -> j=0..5.
    if (lane < 16) {
#pragma unroll
        for (int j = 0; j < VIEWS; ++j) {
            atomicAdd(&sums[j * C + col],  accS[j]);
            atomicAdd(&sumsq[j * C + col], accQ[j]);
        }
    }
}

// ---------------- finalize: V x C partials -> scalar loss ----------------
__global__ void ml_finalize_kernel(const float* __restrict__ sums,
                                   const float* __restrict__ sumsq,
                                   const int* __restrict__ views,
                                   const float* __restrict__ center_mean,
                                   const float* __restrict__ center_std,
                                   float* __restrict__ out,
                                   int N, int C) {
    __shared__ int   cnt[8];
    __shared__ float red[256];
    const int tid = threadIdx.x;

    if (tid < 8) cnt[tid] = 0;
    __syncthreads();
    for (int i = tid; i < N; i += blockDim.x) {
        int v = views[i];
        if ((unsigned)v < (unsigned)VIEWS) atomicAdd(&cnt[v], 1);
    }
    __syncthreads();

    float count[VIEWS];
    bool  valid[VIEWS];
    float nvalid = 0.0f;
#pragma unroll
    for (int v = 0; v < VIEWS; ++v) {
        count[v] = (float)cnt[v];
        valid[v] = count[v] > 1.5f;
        if (valid[v]) nvalid += 1.0f;
    }
    const float moment = (float)N / 10000.0f;

    float partial = 0.0f;
    for (int c = tid; c < C; c += blockDim.x) {
        float mean[VIEWS], stdv[VIEWS];
        float nm = 0.0f, ns = 0.0f;
#pragma unroll
        for (int v = 0; v < VIEWS; ++v) {
            const float n  = fmaxf(count[v], 1.0f);
            const float mu = sums[v * C + c] / n;
            float ss = sumsq[v * C + c] - count[v] * mu * mu;  // = sum((x-mu)^2)
            ss = fmaxf(ss, 0.0f);
            const float denom = fmaxf(count[v] - 1.0f, 1.0f); // unbiased (ddof=1)
            const float sd = sqrtf(ss / denom);
            mean[v] = mu;
            stdv[v] = sd;
            if (valid[v]) { nm += mu; ns += sd; }
        }
        nm /= nvalid;
        ns /= nvalid;
        const float cm = center_mean[c] * (1.0f - moment) + nm * moment;
        const float cs = center_std[c]  * (1.0f - moment) + ns * moment;
#pragma unroll
        for (int v = 0; v < VIEWS; ++v) {
            if (valid[v]) {
                const float dm = mean[v] - cm;
                const float ds = stdv[v] - cs;
                partial += dm * dm + ds * ds;
            }
        }
    }

    red[tid] = partial;
    __syncthreads();
    for (int s = 128; s > 0; s >>= 1) {
        if (tid < s) red[tid] += red[tid + s];
        __syncthreads();
    }
    if (tid == 0) out[0] = red[0] / (2.0f * nvalid);
}

extern "C" void kernel_launch(void* const* d_in, const int* in_sizes, int n_in,
                              void* d_out, int out_size, void* d_ws, size_t ws_size,
                              hipStream_t stream) {
    const float* x     = (const float*)d_in[0];   // (N, C) fp32
    const float* cmean = (const float*)d_in[1];   // (C,)
    const float* cstd  = (const float*)d_in[2];   // (C,)
    const int*   views = (const int*)d_in[3];     // (N,) int32

    const int C = in_sizes[1];
    const int N = in_sizes[3];

    float* sums  = (float*)d_ws;                  // (VIEWS, C)
    float* sumsq = sums + (size_t)VIEWS * (size_t)C;

    // 1) zero the partial-sum workspace (ws is poisoned between runs)
    const int zn = 2 * VIEWS * C;
    ml_zero_kernel<<<(zn + 255) / 256, 256, 0, stream>>>(sums, zn);

    // 2) streaming WMMA segment-sum pass: 2048 waves over the 128MB matrix
    const int col_tiles   = C >> 4;                     // 128
    const int total_waves = col_tiles * ROW_CHUNKS;     // 2048
    const int threads     = total_waves * 32;           // 65536
    ml_seg_wmma_kernel<<<threads / 256, 256, 0, stream>>>(x, views, sums, sumsq, N, C);

    // 3) tiny epilogue: counts, means/stds, EMA centers, scalar loss
    ml_finalize_kernel<<<1, 256, 0, stream>>>(sums, sumsq, views, cmean, cstd,
                                              (float*)d_out, N, C);
}